// AttentionMixer_944892805258
// MI455X (gfx1250) — compile-verified
//
#include <hip/hip_runtime.h>
#include <math.h>

// Problem constants (match reference shapes)
#define S_LEN 200
#define S_PAD 208          // 13 * 16
#define HID   128
#define NHEAD 2
#define NLEV  7
#define NCH   14           // NHEAD * NLEV
#define EST   132          // emb LDS row stride (pad to dodge bank conflicts)
#define NTHREADS 256

typedef __attribute__((ext_vector_type(2))) float v2f;
typedef __attribute__((ext_vector_type(8))) float v8f;

// LDS layout (floats)
#define OFF_EMB   0
#define OFF_QK    (OFF_EMB  + S_PAD * EST)       // 16 x EST
#define OFF_BUFA  (OFF_QK   + 16 * EST)          // 16 x EST (level_emb, then q)
#define OFF_BUFB  (OFF_BUFA + 16 * EST)          // 16 x EST (queries)
#define OFF_ALPHA (OFF_BUFB + 16 * EST)          // 16 x S_PAD, channel-major [n][s]
#define OFF_QB    (OFF_ALPHA + 16 * S_PAD)       // 16
#define OFF_POOL  (OFF_QB   + 16)                // 2 x S_PAD
#define OFF_MASK  (OFF_POOL + 2 * S_PAD)         // S_PAD
#define SMEM_FLOATS (OFF_MASK + S_PAD)

__device__ inline float waveMax(float v) {
    for (int o = 16; o > 0; o >>= 1) v = fmaxf(v, __shfl_xor(v, o, 32));
    return v;
}
__device__ inline float waveSum(float v) {
    for (int o = 16; o > 0; o >>= 1) v += __shfl_xor(v, o, 32);
    return v;
}

// One 16x16 f32 output tile, K = 4*K4, via V_WMMA_F32_16X16X4_F32.
// loadA(m,k) -> A[m][k];  loadB(k,n) -> B[k][n]  (n passed as lane&15)
template <typename FA, typename FB>
__device__ inline v8f wmma_tile_f32(int K4, FA loadA, FB loadB) {
    const int lane = threadIdx.x & 31;
    const int mr   = lane & 15;
    const int koff = (lane < 16) ? 0 : 2;
    v8f c = {0.f, 0.f, 0.f, 0.f, 0.f, 0.f, 0.f, 0.f};
    for (int kk = 0; kk < K4; ++kk) {
        const int k0 = kk * 4 + koff;
        v2f a; a.x = loadA(mr, k0); a.y = loadA(mr, k0 + 1);
        v2f b; b.x = loadB(k0, mr); b.y = loadB(k0 + 1, mr);
        c = __builtin_amdgcn_wmma_f32_16x16x4_f32(false, a, false, b,
                                                  (short)0, c, false, false);
    }
    return c;
}

__global__ __launch_bounds__(NTHREADS)
void attention_mixer_fused(const int*   __restrict__ item_seq,
                           const float* __restrict__ emb,
                           const int*   __restrict__ seq_len,
                           const float* __restrict__ W_lq,
                           const float* __restrict__ b_lq,
                           const float* __restrict__ Wq,
                           const float* __restrict__ bq,
                           const float* __restrict__ Wk,
                           const float* __restrict__ bk,
                           float*       __restrict__ out)
{
    extern __shared__ float smem[];
    float* sEmb   = smem + OFF_EMB;
    float* sQK    = smem + OFF_QK;
    float* sBufA  = smem + OFF_BUFA;
    float* sBufB  = smem + OFF_BUFB;
    float* sAlpha = smem + OFF_ALPHA;
    float* sQB    = smem + OFF_QB;
    float* sPool  = smem + OFF_POOL;
    float* sMask  = smem + OFF_MASK;

    const int b    = blockIdx.x;
    const int tid  = threadIdx.x;
    const int wave = tid >> 5;
    const int lane = tid & 31;
    const int nsub = lane & 15;
    const int rhi  = (lane < 16) ? 0 : 8;

    // ---- Stage 0: load emb[b] tile into LDS, zero pads / scratch ----
    {
        const float4* g = (const float4*)(emb + (size_t)b * S_LEN * HID);
        for (int i = tid; i < S_LEN * (HID / 4); i += NTHREADS) {
            const int row = i >> 5, c4 = i & 31;
            float4 v = g[row * 32 + c4];
            float* dst = &sEmb[row * EST + c4 * 4];
            dst[0] = v.x; dst[1] = v.y; dst[2] = v.z; dst[3] = v.w;
        }
        for (int i = tid; i < (S_PAD - S_LEN) * EST; i += NTHREADS)
            sEmb[S_LEN * EST + i] = 0.f;
        for (int i = tid; i < 16 * EST; i += NTHREADS) { sBufA[i] = 0.f; sBufB[i] = 0.f; }
        for (int i = tid; i < 2 * EST; i += NTHREADS) sQK[14 * EST + i] = 0.f;
        if (tid < 16) sQB[tid] = 0.f;
        for (int s = tid; s < S_PAD; s += NTHREADS)
            sMask[s] = (s < S_LEN && item_seq[(size_t)b * S_LEN + s] > 0) ? 1.f : 0.f;
    }
    const int len = seq_len[b];
    __syncthreads();

    // ---- Stage 1: multi-level gather + cumsum -> sBufA rows 0..6 ----
    if (tid < HID) {
        const int c = tid;
        float acc = 0.f;
        for (int j = 0; j < NLEV; ++j) {
            const int t   = len - (j + 1);
            const int idx = (t >= 0) ? t : (S_LEN - 1);   // clamp(-1)+S semantics
            acc += sEmb[idx * EST + c];
            sBufA[j * EST + c] = acc;
        }
    }
    __syncthreads();

    // ---- Stage 2: queries = level_emb @ W_lq^T + b_lq  -> sBufB ----
    {
        v8f c = wmma_tile_f32(HID / 4,
            [&](int m, int k) { return sBufA[m * EST + k]; },
            [&](int k, int n) { return W_lq[(wave * 16 + n) * HID + k]; });
        const int col = wave * 16 + nsub;
        const float bias = b_lq[col];
        for (int r = 0; r < 8; ++r) {
            const int m = r + rhi;
            if (m < NLEV) sBufB[m * EST + col] = c[r] + bias;
        }
    }
    __syncthreads();

    // ---- Stage 3: q = queries @ Wq^T + bq  -> sBufA ----
    {
        v8f c = wmma_tile_f32(HID / 4,
            [&](int m, int k) { return sBufB[m * EST + k]; },
            [&](int k, int n) { return Wq[(wave * 16 + n) * HID + k]; });
        const int col = wave * 16 + nsub;
        const float bias = bq[col];
        for (int r = 0; r < 8; ++r) {
            const int m = r + rhi;
            if (m < NLEV) sBufA[m * EST + col] = c[r] + bias;
        }
    }
    __syncthreads();

    // ---- Stage 4: qk[h*7+l][:] = q_h[l][:] @ Wk[h*64:(h+1)*64][:]  (folds k-proj) ----
    for (int h = 0; h < NHEAD; ++h) {
        v8f c = wmma_tile_f32(64 / 4,
            [&](int m, int k) { return sBufA[m * EST + h * 64 + k]; },
            [&](int k, int n) { return Wk[(h * 64 + k) * HID + (wave * 16 + n)]; });
        const int col = wave * 16 + nsub;
        for (int r = 0; r < 8; ++r) {
            const int m = r + rhi;
            if (m < NLEV) sQK[(h * NLEV + m) * EST + col] = c[r];
        }
    }
    if (tid < NCH) {          // qb[n] = q_h[l] . bk_h   (folded k bias)
        const int h = tid / NLEV, l = tid % NLEV;
        float s = 0.f;
        for (int d2 = 0; d2 < 64; ++d2)
            s += sBufA[l * EST + h * 64 + d2] * bk[h * 64 + d2];
        sQB[tid] = s;
    }
    __syncthreads();

    // ---- Stage 5: alpha[n][s] = sigmoid(emb[s,:].qk[n,:] + qb[n]) via WMMA ----
    for (int t = wave; t < S_PAD / 16; t += 8) {
        v8f c = wmma_tile_f32(HID / 4,
            [&](int m, int k) { return sEmb[(t * 16 + m) * EST + k]; },
            [&](int k, int n) { return sQK[n * EST + k]; });
        const float bias = sQB[nsub];
        for (int r = 0; r < 8; ++r) {
            const int m = t * 16 + r + rhi;
            const float x = c[r] + bias;
            sAlpha[nsub * S_PAD + m] = 1.f / (1.f + __expf(-x));
        }
    }
    __syncthreads();

    // ---- Stage 6: softmax over s (per channel, unmasked) ----
    for (int n = wave; n < NCH; n += 8) {
        float mx = -3.0e38f;
        for (int s = lane; s < S_LEN; s += 32) mx = fmaxf(mx, sAlpha[n * S_PAD + s]);
        mx = waveMax(mx);
        float sum = 0.f;
        for (int s = lane; s < S_LEN; s += 32) {
            const float e = __expf(sAlpha[n * S_PAD + s] - mx);
            sAlpha[n * S_PAD + s] = e;
            sum += e;
        }
        sum = waveSum(sum);
        const float inv = 1.f / sum;
        for (int s = lane; s < S_LEN; s += 32) sAlpha[n * S_PAD + s] *= inv;
    }
    __syncthreads();

    // ---- Stage 7: LP-pool (p=4) over levels, mask, softmax over s ----
    if (wave < NHEAD) {
        const int h = wave;
        float mx = -3.0e38f;
        for (int s = lane; s < S_LEN; s += 32) {
            float acc = 0.f;
            for (int l = 0; l < NLEV; ++l) {
                const float a  = sAlpha[(h * NLEV + l) * S_PAD + s];
                const float a2 = a * a;
                acc += a2 * a2;
            }
            const float p = sqrtf(sqrtf(acc));                 // acc^(1/4)
            const float v = (sMask[s] > 0.f) ? p : -3.0e38f;
            sPool[h * S_PAD + s] = v;
            mx = fmaxf(mx, v);
        }
        mx = waveMax(mx);
        float sum = 0.f;
        for (int s = lane; s < S_LEN; s += 32) {
            const float e = __expf(sPool[h * S_PAD + s] - mx);
            sPool[h * S_PAD + s] = e;
            sum += e;
        }
        sum = waveSum(sum);
        const float inv = 1.f / sum;
        for (int s = lane; s < S_LEN; s += 32) sPool[h * S_PAD + s] *= inv;
    }
    __syncthreads();

    // ---- Stage 8: out[b][j] = sum_s w[s][j/64] * emb[s][j] ----
    if (tid < HID) {
        const int j = tid, h = j >> 6;
        float acc = 0.f;
        for (int s = 0; s < S_LEN; ++s)
            acc += sPool[h * S_PAD + s] * sEmb[s * EST + j];
        out[(size_t)b * HID + j] = acc;
    }
}

extern "C" void kernel_launch(void* const* d_in, const int* in_sizes, int n_in,
                              void* d_out, int out_size, void* d_ws, size_t ws_size,
                              hipStream_t stream) {
    const int*   item_seq = (const int*)  d_in[0];
    const float* emb      = (const float*)d_in[1];
    const int*   lens     = (const int*)  d_in[2];
    const float* W_lq     = (const float*)d_in[3];
    const float* b_lq     = (const float*)d_in[4];
    const float* Wq       = (const float*)d_in[5];
    const float* bq       = (const float*)d_in[6];
    const float* Wk       = (const float*)d_in[7];
    const float* bk       = (const float*)d_in[8];
    float* out = (float*)d_out;

    const int B = in_sizes[2];                     // batch = #seq lens
    const size_t smem_bytes = (size_t)SMEM_FLOATS * sizeof(float);

    attention_mixer_fused<<<dim3(B), dim3(NTHREADS), smem_bytes, stream>>>(
        item_seq, emb, lens, W_lq, b_lq, Wq, bq, Wk, bk, out);
}